// BaseUNet_70600672411683
// MI455X (gfx1250) — compile-verified
//
#include <hip/hip_runtime.h>
#include <hip/hip_bf16.h>

typedef __attribute__((ext_vector_type(2))) float v2f;
typedef __attribute__((ext_vector_type(8))) float v8f;

#define THREADS 256

static inline int cdiv(long long a, long long b) { return (int)((a + b - 1) / b); }

// ---------------------------------------------------------------- fill / copy
__global__ void k_fill_f(float* p, float v, long long n) {
  long long i = (long long)blockIdx.x * blockDim.x + threadIdx.x;
  if (i < n) p[i] = v;
}
__global__ void k_fill_i(int* p, int v, long long n) {
  long long i = (long long)blockIdx.x * blockDim.x + threadIdx.x;
  if (i < n) p[i] = v;
}
__global__ void k_fill_u(unsigned* p, unsigned v, long long n) {
  long long i = (long long)blockIdx.x * blockDim.x + threadIdx.x;
  if (i < n) p[i] = v;
}
__global__ void k_copy_f(float* d, const float* s, long long n) {
  long long i = (long long)blockIdx.x * blockDim.x + threadIdx.x;
  if (i < n) d[i] = s[i];
}

// ---------------------------------------------------------------- WMMA GEMM
// C[M,Co] = A[M,K] @ W[K,Co] (+bias) (+relu), all row-major fp32.
// One wave computes a 16 x (16*TN) output stripe: the A 16x4 fragment is
// loaded ONCE per K-step and reused across TN column tiles (TN WMMAs/step),
// cutting A traffic TN-fold and packing the matrix pipe.
// A 16x4 layout: lanes 0-15 hold K=0,1 (2 VGPRs); lanes 16-31 hold K=2,3.
// B 4x16 layout mirrored. C/D: VGPR v holds M=v (lanes 0-15) / M=v+8 (16-31).
template <int TN>
__global__ void k_gemm_wmma(const float* __restrict__ A, const float* __restrict__ W,
                            const float* __restrict__ bias, float* __restrict__ C,
                            int M, int K, int Co, int relu) {
  const int lane = threadIdx.x;
  const int half = lane >> 4;
  const int l16  = lane & 15;
  const int tm = blockIdx.x;
  const int cb = blockIdx.y * (16 * TN);        // column stripe base
  int row  = tm * 16 + l16;
  int rowc = row < M ? row : (M - 1);           // clamp loads; stores guarded
  const float* arow = A + (long long)rowc * K;
  v8f acc[TN];
#pragma unroll
  for (int t = 0; t < TN; ++t) acc[t] = (v8f){0.f,0.f,0.f,0.f,0.f,0.f,0.f,0.f};
  for (int k0 = 0; k0 < K; k0 += 4) {
    int ka = k0 + 2 * half;
    v2f a = *(const v2f*)(arow + ka);           // aligned b64 load, reused TN times
    const float* wrow = W + (long long)ka * Co + cb + l16;
#pragma unroll
    for (int t = 0; t < TN; ++t) {
      v2f b;
      b.x = wrow[t * 16];
      b.y = wrow[(long long)Co + t * 16];
      acc[t] = __builtin_amdgcn_wmma_f32_16x16x4_f32(false, a, false, b,
                                                     (short)0, acc[t], false, false);
    }
  }
#pragma unroll
  for (int t = 0; t < TN; ++t) {
    int col = cb + t * 16 + l16;
    float bv = bias ? bias[col] : 0.0f;
#pragma unroll
    for (int v = 0; v < 8; ++v) {
      int r = tm * 16 + v + 8 * half;
      if (r < M) {
        float o = acc[t][v] + bv;
        if (relu) o = fmaxf(o, 0.0f);
        C[(long long)r * Co + col] = o;
      }
    }
  }
}

// ---------------------------------------------------------------- degree / dinv
__global__ void k_deg_accum(const int* __restrict__ dst, const float* __restrict__ ew,
                            float* deg, int E) {
  int e = blockIdx.x * blockDim.x + threadIdx.x;
  if (e >= E) return;
  float w = ew[e];
  if (w != 0.0f) atomicAdd(&deg[dst[e]], w);
}
__global__ void k_dinv(float* d, int n) {
  int i = blockIdx.x * blockDim.x + threadIdx.x;
  if (i < n) d[i] = 1.0f / sqrtf(d[i] + 2.0f);  // deg >= 2 always (improved self-loop)
}

// ---------------------------------------------------------------- GCN scatter
// Per-edge norm precomputed once (removes C-fold redundant dinv/ew gathers).
__global__ void k_edge_norm(const int* __restrict__ src, const int* __restrict__ dst,
                            const float* __restrict__ ew, const float* __restrict__ dinv,
                            float* nrm, int E) {
  int e = blockIdx.x * blockDim.x + threadIdx.x;
  if (e >= E) return;
  float w = ew[e];
  nrm[e] = (w != 0.0f) ? dinv[src[e]] * w * dinv[dst[e]] : 0.0f;
}
__global__ void k_gcn_aggregate(const int* __restrict__ src, const int* __restrict__ dst,
                                const float* __restrict__ nrm,
                                const float* __restrict__ h, float* agg, int E, int C) {
  long long id = (long long)blockIdx.x * blockDim.x + threadIdx.x;
  long long tot = (long long)E * C;
  if (id >= tot) return;
  int e = (int)(id / C);
  int c = (int)(id % C);
  float w = nrm[e];
  if (w == 0.0f) return;                        // dropped edges contribute nothing
  atomicAdd(&agg[(long long)dst[e] * C + c], h[(long long)src[e] * C + c] * w);
}
__global__ void k_gcn_combine(const float* __restrict__ agg, const float* __restrict__ h,
                              const float* __restrict__ dinv, const float* __restrict__ bias,
                              float* out, long long tot, int C, int relu) {
  long long i = (long long)blockIdx.x * blockDim.x + threadIdx.x;
  if (i >= tot) return;
  int c = (int)(i % C);
  int n = (int)(i / C);
  float dv = dinv[n];
  float o = agg[i] + 2.0f * dv * dv * h[i] + bias[c];
  if (relu) o = fmaxf(o, 0.0f);
  out[i] = o;
}

// ---------------------------------------------------------------- BatchNorm
__global__ void k_bn_reduce(const float* __restrict__ x, float* S, float* SS, int N, int C) {
  int c = threadIdx.x;                           // blockDim.x == C
  int per = (N + gridDim.x - 1) / gridDim.x;
  int r0 = blockIdx.x * per;
  int r1 = r0 + per; if (r1 > N) r1 = N;
  float a = 0.f, q = 0.f;
  for (int r = r0; r < r1; ++r) {
    float v = x[(long long)r * C + c];
    a += v; q += v * v;
  }
  atomicAdd(&S[c], a);
  atomicAdd(&SS[c], q);
}
__global__ void k_bn_finalize(const float* S, const float* SS, const float* g, const float* b,
                              float* scale, float* shift, int N, int C) {
  int c = threadIdx.x;
  if (c >= C) return;
  float mu  = S[c] / (float)N;
  float var = SS[c] / (float)N - mu * mu;        // biased variance
  float rs  = 1.0f / sqrtf(var + 1e-5f);
  float sc  = g[c] * rs;
  scale[c] = sc;
  shift[c] = b[c] - mu * sc;
}
__global__ void k_bn_apply(float* x, const float* scale, const float* shift,
                           long long tot, int C, int relu) {
  long long i = (long long)blockIdx.x * blockDim.x + threadIdx.x;
  if (i >= tot) return;
  int c = (int)(i % C);
  float o = x[i] * scale[c] + shift[c];
  if (relu) o = fmaxf(o, 0.0f);
  x[i] = o;
}

// ---------------------------------------------------------------- TopK pooling
__global__ void k_pnorm(const float* p, float* out, int C) {
  __shared__ float sh[256];
  int i = threadIdx.x;
  sh[i] = (i < C) ? p[i] * p[i] : 0.0f;
  __syncthreads();
  for (int s = 128; s > 0; s >>= 1) { if (i < s) sh[i] += sh[i + s]; __syncthreads(); }
  if (i == 0) out[0] = sqrtf(sh[0]);
}
__global__ void k_score(const float* __restrict__ x, const float* __restrict__ p,
                        const float* pnorm, float* sc, int N, int C) {
  int i = blockIdx.x * blockDim.x + threadIdx.x;
  if (i >= N) return;
  const float* row = x + (long long)i * C;
  float acc = 0.f;
  for (int c = 0; c < C; ++c) acc += row[c] * p[c];
  sc[i] = tanhf(acc / pnorm[0]);
}
__device__ __forceinline__ unsigned fkey(float f) {   // monotone float->uint
  unsigned u = __float_as_uint(f);
  return (u & 0x80000000u) ? ~u : (u | 0x80000000u);
}
// state: [0]=prefix [1]=mask [2]=k_remaining/need [3]=pad [4]=cntGT [5]=cntEQ
__global__ void k_rs_init(unsigned* st, unsigned k) {
  if (threadIdx.x == 0) { st[0]=0u; st[1]=0u; st[2]=k; st[3]=0u; st[4]=0u; st[5]=0u; }
}
__global__ void k_rs_hist(const float* sc, int n, unsigned* hist, const unsigned* st, int shift) {
  int i = blockIdx.x * blockDim.x + threadIdx.x;
  if (i >= n) return;
  unsigned key = fkey(sc[i]);
  if ((key & st[1]) == st[0]) atomicAdd(&hist[(key >> shift) & 255u], 1u);
}
__global__ void k_rs_scan(const unsigned* hist, unsigned* st, int shift) {
  if (threadIdx.x || blockIdx.x) return;
  unsigned k = st[2];
  for (int d = 255; d >= 0; --d) {
    unsigned c = hist[d];
    if (k <= c) { st[0] |= ((unsigned)d) << shift; st[1] |= (255u << shift); st[2] = k; return; }
    k -= c;
  }
}
__global__ void k_rs_compact(const float* sc, int n, unsigned* st, int* perm, int k) {
  int i = blockIdx.x * blockDim.x + threadIdx.x;
  if (i >= n) return;
  unsigned key  = fkey(sc[i]);
  unsigned T    = st[0];
  unsigned need = st[2];
  if (key > T) {
    unsigned p = atomicAdd(&st[4], 1u);
    perm[p] = i;
  } else if (key == T) {
    unsigned s = atomicAdd(&st[5], 1u);
    if (s < need) perm[(unsigned)k - need + s] = i;
  }
}
__global__ void k_pool_gather(const float* __restrict__ x, const int* __restrict__ perm,
                              const float* __restrict__ sc, float* xk, long long tot, int C) {
  long long id = (long long)blockIdx.x * blockDim.x + threadIdx.x;
  if (id >= tot) return;
  int j = (int)(id / C), c = (int)(id % C);
  int r = perm[j];
  xk[id] = x[(long long)r * C + c] * sc[r];
}
__global__ void k_set_nidx(const int* perm, int* nidx, int k) {
  int j = blockIdx.x * blockDim.x + threadIdx.x;
  if (j < k) nidx[perm[j]] = j;
}
__global__ void k_remap(const int* __restrict__ src, const int* __restrict__ dst,
                        const float* __restrict__ ew, const int* __restrict__ nidx,
                        int* s_o, int* d_o, float* w_o, int E) {
  int e = blockIdx.x * blockDim.x + threadIdx.x;
  if (e >= E) return;
  int ns = nidx[src[e]], nd = nidx[dst[e]];
  bool keep = (ns >= 0) && (nd >= 0);
  s_o[e] = keep ? ns : 0;
  d_o[e] = keep ? nd : 0;
  w_o[e] = keep ? ew[e] : 0.0f;
}
__global__ void k_unpool(const float* __restrict__ d, const int* __restrict__ perm,
                         float* xu, long long tot, int C) {
  long long id = (long long)blockIdx.x * blockDim.x + threadIdx.x;
  if (id >= tot) return;
  int j = (int)(id / C), c = (int)(id % C);
  long long t = (long long)perm[j] * C + c;
  xu[t] = (d[id] + xu[t]) * 0.5f;
}

// ================================================================ launch
extern "C" void kernel_launch(void* const* d_in, const int* in_sizes, int n_in,
                              void* d_out, int out_size, void* d_ws, size_t ws_size,
                              hipStream_t stream) {
  (void)n_in; (void)out_size; (void)ws_size;
  const float* x        = (const float*)d_in[0];
  const int*   esrc     = (const int*)d_in[1];
  const int*   edst     = (const int*)d_in[2];
  const float* e0_mlp_W = (const float*)d_in[3];
  const float* e0_mlp_b = (const float*)d_in[4];
  const float* e0_bn1_g = (const float*)d_in[5];
  const float* e0_bn1_b = (const float*)d_in[6];
  const float* e0_conv_W= (const float*)d_in[7];
  const float* e0_conv_b= (const float*)d_in[8];
  const float* e0_n_g   = (const float*)d_in[9];
  const float* e0_n_b   = (const float*)d_in[10];
  const float* e1_p     = (const float*)d_in[11];
  const float* e1_mlp_W = (const float*)d_in[12];
  const float* e1_mlp_b = (const float*)d_in[13];
  const float* e1_bn1_g = (const float*)d_in[14];
  const float* e1_bn1_b = (const float*)d_in[15];
  const float* e1_conv_W= (const float*)d_in[16];
  const float* e1_conv_b= (const float*)d_in[17];
  const float* e1_n_g   = (const float*)d_in[18];
  const float* e1_n_b   = (const float*)d_in[19];
  const float* e2_p     = (const float*)d_in[20];
  const float* e2_mlp_W = (const float*)d_in[21];
  const float* e2_mlp_b = (const float*)d_in[22];
  const float* e2_bn1_g = (const float*)d_in[23];
  const float* e2_bn1_b = (const float*)d_in[24];
  const float* e2_conv_W= (const float*)d_in[25];
  const float* e2_conv_b= (const float*)d_in[26];
  const float* e2_n_g   = (const float*)d_in[27];
  const float* e2_n_b   = (const float*)d_in[28];
  const float* d2_W     = (const float*)d_in[29];
  const float* d2_b     = (const float*)d_in[30];
  const float* d1_W     = (const float*)d_in[31];
  const float* d1_b     = (const float*)d_in[32];
  const float* d0_W     = (const float*)d_in[33];
  const float* d0_b     = (const float*)d_in[34];

  const int C0 = 16, C1 = 64, C2 = 128, C3 = 256;
  const int N  = in_sizes[0] / C0;
  const int E  = in_sizes[1];
  const int k1 = (N + 1) / 2;      // ceil(0.5*N)
  const int k2 = (k1 + 1) / 2;

  // ---- workspace bump allocator ----
  char* wsb = (char*)d_ws;
  size_t off = 0;
  auto alloc = [&](size_t bytes) -> void* {
    void* p = wsb + off;
    off += (bytes + 255) & ~(size_t)255;
    return p;
  };
  float* ones   = (float*)alloc((size_t)E * 4);
  float* enorm  = (float*)alloc((size_t)E * 4);
  float* dinv0  = (float*)alloc((size_t)N * 4);
  float* dinv1  = (float*)alloc((size_t)k1 * 4);
  float* dinv2  = (float*)alloc((size_t)k2 * 4);
  float* x1     = (float*)alloc((size_t)N * C1 * 4);
  float* x2     = (float*)alloc((size_t)k1 * C2 * 4);
  float* x3     = (float*)alloc((size_t)k2 * C3 * 4);
  size_t bufE = (size_t)N * C1;
  if ((size_t)k1 * C2 > bufE) bufE = (size_t)k1 * C2;
  if ((size_t)k2 * C3 > bufE) bufE = (size_t)k2 * C3;
  float* tbuf   = (float*)alloc(bufE * 4);
  float* hbuf   = (float*)alloc(bufE * 4);
  float* abuf   = (float*)alloc(bufE * 4);
  float* xp1    = (float*)alloc((size_t)k1 * C1 * 4);
  float* xp2    = (float*)alloc((size_t)k2 * C2 * 4);
  float* xu2    = (float*)alloc((size_t)k1 * C2 * 4);
  float* xu1    = (float*)alloc((size_t)N * C1 * 4);
  float* score1 = (float*)alloc((size_t)N * 4);
  float* score2 = (float*)alloc((size_t)k1 * 4);
  int*   perm1  = (int*)alloc((size_t)k1 * 4);
  int*   perm2  = (int*)alloc((size_t)k2 * 4);
  int*   nidx1  = (int*)alloc((size_t)N * 4);
  int*   nidx2  = (int*)alloc((size_t)k1 * 4);
  int*   s1 = (int*)alloc((size_t)E * 4);
  int*   t1 = (int*)alloc((size_t)E * 4);
  float* ew1 = (float*)alloc((size_t)E * 4);
  int*   s2 = (int*)alloc((size_t)E * 4);
  int*   t2 = (int*)alloc((size_t)E * 4);
  float* ew2 = (float*)alloc((size_t)E * 4);
  float* bnstats = (float*)alloc(1024 * 4);   // S | SS | scale | shift (256 each)
  float* pnorm   = (float*)alloc(256);
  unsigned* rsbuf = (unsigned*)alloc((256 + 8) * 4);  // hist | state

  // ---- helpers ----
  auto fillf = [&](float* p, float v, long long n) {
    k_fill_f<<<cdiv(n, THREADS), THREADS, 0, stream>>>(p, v, n);
  };
  auto filli = [&](int* p, int v, long long n) {
    k_fill_i<<<cdiv(n, THREADS), THREADS, 0, stream>>>(p, v, n);
  };
  auto copyf = [&](float* d, const float* s, long long n) {
    k_copy_f<<<cdiv(n, THREADS), THREADS, 0, stream>>>(d, s, n);
  };
  auto gemm = [&](const float* A, const float* W, const float* bias, float* Cm,
                  int M, int K, int Co, int relu) {
    if (Co % 64 == 0) {          // 16x64 stripe per wave: A reused across 4 WMMAs
      dim3 g(cdiv(M, 16), Co / 64), b(32, 1, 1);
      k_gemm_wmma<4><<<g, b, 0, stream>>>(A, W, bias, Cm, M, K, Co, relu);
    } else {                     // Co == 16
      dim3 g(cdiv(M, 16), Co / 16), b(32, 1, 1);
      k_gemm_wmma<1><<<g, b, 0, stream>>>(A, W, bias, Cm, M, K, Co, relu);
    }
  };
  auto bn = [&](float* xio, const float* g, const float* b, int Nn, int Cc, int relu) {
    float* S = bnstats; float* SS = bnstats + 256;
    float* sc = bnstats + 512; float* sh = bnstats + 768;
    k_fill_f<<<2, THREADS, 0, stream>>>(bnstats, 0.0f, 512);
    k_bn_reduce<<<256, Cc, 0, stream>>>(xio, S, SS, Nn, Cc);
    k_bn_finalize<<<1, Cc, 0, stream>>>(S, SS, g, b, sc, sh, Nn, Cc);
    long long tot = (long long)Nn * Cc;
    k_bn_apply<<<cdiv(tot, THREADS), THREADS, 0, stream>>>(xio, sc, sh, tot, Cc, relu);
  };
  auto gcn = [&](const float* xin, const float* W, const float* bias,
                 const int* src, const int* dst, const float* ew, const float* dinv,
                 float* out, int Nn, int K, int Co, int relu) {
    gemm(xin, W, nullptr, hbuf, Nn, K, Co, 0);
    k_edge_norm<<<cdiv(E, THREADS), THREADS, 0, stream>>>(src, dst, ew, dinv, enorm, E);
    fillf(abuf, 0.0f, (long long)Nn * Co);
    long long tot = (long long)E * Co;
    k_gcn_aggregate<<<cdiv(tot, THREADS), THREADS, 0, stream>>>(src, dst, enorm, hbuf, abuf, E, Co);
    long long nt = (long long)Nn * Co;
    k_gcn_combine<<<cdiv(nt, THREADS), THREADS, 0, stream>>>(abuf, hbuf, dinv, bias, out, nt, Co, relu);
  };
  auto degprep = [&](const int* dst, const float* ew, float* dinv, int Nn) {
    fillf(dinv, 0.0f, Nn);
    k_deg_accum<<<cdiv(E, THREADS), THREADS, 0, stream>>>(dst, ew, dinv, E);
    k_dinv<<<cdiv(Nn, THREADS), THREADS, 0, stream>>>(dinv, Nn);
  };
  auto topk = [&](const float* sc, int n, int k, int* perm) {
    k_rs_init<<<1, 32, 0, stream>>>(rsbuf + 256, (unsigned)k);
    for (int shift = 24; shift >= 0; shift -= 8) {
      k_fill_u<<<1, 256, 0, stream>>>(rsbuf, 0u, 256);
      k_rs_hist<<<cdiv(n, THREADS), THREADS, 0, stream>>>(sc, n, rsbuf, rsbuf + 256, shift);
      k_rs_scan<<<1, 32, 0, stream>>>(rsbuf, rsbuf + 256, shift);
    }
    k_rs_compact<<<cdiv(n, THREADS), THREADS, 0, stream>>>(sc, n, rsbuf + 256, perm, k);
  };

  // ---- level-0 graph prep ----
  fillf(ones, 1.0f, E);
  degprep(edst, ones, dinv0, N);

  // ---- encoder0 ----
  gemm(x, e0_mlp_W, e0_mlp_b, tbuf, N, C0, C1, 1);
  bn(tbuf, e0_bn1_g, e0_bn1_b, N, C1, 0);
  gcn(tbuf, e0_conv_W, e0_conv_b, esrc, edst, ones, dinv0, x1, N, C1, C1, 0);
  bn(x1, e0_n_g, e0_n_b, N, C1, 1);

  // ---- pool1 ----
  k_pnorm<<<1, 256, 0, stream>>>(e1_p, pnorm, C1);
  k_score<<<cdiv(N, THREADS), THREADS, 0, stream>>>(x1, e1_p, pnorm, score1, N, C1);
  topk(score1, N, k1, perm1);
  {
    long long tot = (long long)k1 * C1;
    k_pool_gather<<<cdiv(tot, THREADS), THREADS, 0, stream>>>(x1, perm1, score1, xp1, tot, C1);
  }
  filli(nidx1, -1, N);
  k_set_nidx<<<cdiv(k1, THREADS), THREADS, 0, stream>>>(perm1, nidx1, k1);
  k_remap<<<cdiv(E, THREADS), THREADS, 0, stream>>>(esrc, edst, ones, nidx1, s1, t1, ew1, E);
  degprep(t1, ew1, dinv1, k1);

  // ---- encoder1 ----
  gemm(xp1, e1_mlp_W, e1_mlp_b, tbuf, k1, C1, C2, 1);
  bn(tbuf, e1_bn1_g, e1_bn1_b, k1, C2, 0);
  gcn(tbuf, e1_conv_W, e1_conv_b, s1, t1, ew1, dinv1, x2, k1, C2, C2, 0);
  bn(x2, e1_n_g, e1_n_b, k1, C2, 1);

  // ---- pool2 ----
  k_pnorm<<<1, 256, 0, stream>>>(e2_p, pnorm, C2);
  k_score<<<cdiv(k1, THREADS), THREADS, 0, stream>>>(x2, e2_p, pnorm, score2, k1, C2);
  topk(score2, k1, k2, perm2);
  {
    long long tot = (long long)k2 * C2;
    k_pool_gather<<<cdiv(tot, THREADS), THREADS, 0, stream>>>(x2, perm2, score2, xp2, tot, C2);
  }
  filli(nidx2, -1, k1);
  k_set_nidx<<<cdiv(k2, THREADS), THREADS, 0, stream>>>(perm2, nidx2, k2);
  k_remap<<<cdiv(E, THREADS), THREADS, 0, stream>>>(s1, t1, ew1, nidx2, s2, t2, ew2, E);
  degprep(t2, ew2, dinv2, k2);

  // ---- encoder2 ----
  gemm(xp2, e2_mlp_W, e2_mlp_b, tbuf, k2, C2, C3, 1);
  bn(tbuf, e2_bn1_g, e2_bn1_b, k2, C3, 0);
  gcn(tbuf, e2_conv_W, e2_conv_b, s2, t2, ew2, dinv2, x3, k2, C3, C3, 0);
  bn(x3, e2_n_g, e2_n_b, k2, C3, 1);

  // ---- decoder2 + unpool into x2 ----
  gcn(x3, d2_W, d2_b, s2, t2, ew2, dinv2, tbuf, k2, C3, C2, 1);
  copyf(xu2, x2, (long long)k1 * C2);
  {
    long long tot = (long long)k2 * C2;
    k_unpool<<<cdiv(tot, THREADS), THREADS, 0, stream>>>(tbuf, perm2, xu2, tot, C2);
  }

  // ---- decoder1 + unpool into x1 ----
  gcn(xu2, d1_W, d1_b, s1, t1, ew1, dinv1, tbuf, k1, C2, C1, 1);
  copyf(xu1, x1, (long long)N * C1);
  {
    long long tot = (long long)k1 * C1;
    k_unpool<<<cdiv(tot, THREADS), THREADS, 0, stream>>>(tbuf, perm1, xu1, tot, C1);
  }

  // ---- decoder0 -> output ----
  gcn(xu1, d0_W, d0_b, esrc, edst, ones, dinv0, (float*)d_out, N, C1, C0, 0);
}